// GPRGNN_70188355551840
// MI455X (gfx1250) — compile-verified
//
#include <hip/hip_runtime.h>
#include <hip/hip_bf16.h>

typedef __attribute__((ext_vector_type(2))) float v2f;
typedef __attribute__((ext_vector_type(8))) float v8f;

#define NFEAT 512
#define NHID  256
#define NCLS  64
#define KHOPS 10

// ---------------------------------------------------------------------------
// Fused MLP: h_out[n,64] = relu(x@W1 + b1) @ W2 + b2, via fp32 WMMA 16x16x4.
// Block = 256 threads (8 waves), 16 rows per block.
//   Phase 1: stage 16x512 x-tile in LDS; 8 waves x 2 col-tiles -> 16x256 h1
//            (relu + b1 applied) into LDS.
//   Phase 2: waves 0-3 compute the 16x64 output from the LDS h1 tile.
// ---------------------------------------------------------------------------
__global__ __launch_bounds__(256) void gpr_mlp_wmma(
    const float* __restrict__ x, const float* __restrict__ W1,
    const float* __restrict__ b1, const float* __restrict__ W2,
    const float* __restrict__ b2, float* __restrict__ h_out, int n) {
  __shared__ float sx[16 * NFEAT];  // 32 KB
  __shared__ float sh[16 * NHID];   // 16 KB
  const int tid  = threadIdx.x;
  const int row0 = blockIdx.x * 16;

  // Stage x tile (zero-pad rows >= n) with float4 loads.
  for (int i = tid; i < 16 * NFEAT / 4; i += 256) {
    int r = (i * 4) / NFEAT;
    int c = (i * 4) % NFEAT;
    int gr = row0 + r;
    float4 v = (gr < n) ? *(const float4*)(x + (size_t)gr * NFEAT + c)
                        : make_float4(0.f, 0.f, 0.f, 0.f);
    *(float4*)(sx + r * NFEAT + c) = v;
  }
  __syncthreads();

  const int wave = tid >> 5;
  const int lane = tid & 31;
  const int m    = lane & 15;   // row (A) / col (B,C) within tile
  const int kp   = lane >> 4;   // K-pair select (0 -> K{0,1}, 1 -> K{2,3})

  // ---- Phase 1: h1 = relu(x@W1 + b1) ----
  for (int t = 0; t < 2; ++t) {
    const int col0 = (wave * 2 + t) * 16;
    v8f acc = {};
    for (int k = 0; k < NFEAT; k += 4) {
      v2f a, b;
      a.x = sx[m * NFEAT + k + 2 * kp];
      a.y = sx[m * NFEAT + k + 2 * kp + 1];
      b.x = W1[(size_t)(k + 2 * kp) * NHID + col0 + m];
      b.y = W1[(size_t)(k + 2 * kp + 1) * NHID + col0 + m];
      acc = __builtin_amdgcn_wmma_f32_16x16x4_f32(
          false, a, false, b, (short)0, acc, false, false);
    }
    const float bias = b1[col0 + m];
    const int mBase = kp * 8;
#pragma unroll
    for (int v = 0; v < 8; ++v) {
      sh[(mBase + v) * NHID + col0 + m] = fmaxf(acc[v] + bias, 0.0f);
    }
  }
  __syncthreads();

  // ---- Phase 2: out = h1@W2 + b2 (waves 0-3 cover 64 output cols) ----
  if (wave < 4) {
    const int col0 = wave * 16;
    v8f acc = {};
    for (int k = 0; k < NHID; k += 4) {
      v2f a, b;
      a.x = sh[m * NHID + k + 2 * kp];
      a.y = sh[m * NHID + k + 2 * kp + 1];
      b.x = W2[(size_t)(k + 2 * kp) * NCLS + col0 + m];
      b.y = W2[(size_t)(k + 2 * kp + 1) * NCLS + col0 + m];
      acc = __builtin_amdgcn_wmma_f32_16x16x4_f32(
          false, a, false, b, (short)0, acc, false, false);
    }
    const float bias = b2[col0 + m];
    const int mBase = kp * 8;
#pragma unroll
    for (int v = 0; v < 8; ++v) {
      int gr = row0 + mBase + v;
      if (gr < n) h_out[(size_t)gr * NCLS + col0 + m] = acc[v] + bias;
    }
  }
}

// ---------------------------------------------------------------------------
// GCN normalization: deg (with self loop) then dinv = rsqrt(deg) in place.
// ---------------------------------------------------------------------------
__global__ __launch_bounds__(256) void deg_init(float* __restrict__ deg, int n) {
  int i = blockIdx.x * 256 + threadIdx.x;
  if (i < n) deg[i] = 1.0f;  // self loop
}

__global__ __launch_bounds__(256) void deg_accum(const long long* __restrict__ dst,
                                                 float* __restrict__ deg, int e) {
  int i = blockIdx.x * 256 + threadIdx.x;
  if (i < e) atomicAdd(&deg[(size_t)dst[i]], 1.0f);
}

__global__ __launch_bounds__(256) void deg_to_dinv(float* __restrict__ deg, int n) {
  int i = blockIdx.x * 256 + threadIdx.x;
  if (i < n) deg[i] = rsqrtf(deg[i]);  // deg >= 1 always (self loop)
}

// h_out = dinv^2 * h_in   (self-loop contribution = hop init)
__global__ __launch_bounds__(256) void hop_init(const float* __restrict__ h_in,
                                                const float* __restrict__ dinv,
                                                float* __restrict__ h_out, int n) {
  int idx = blockIdx.x * 256 + threadIdx.x;
  if (idx < n * NCLS) {
    float d = dinv[idx >> 6];
    h_out[idx] = d * d * h_in[idx];
  }
}

// Edge scatter: h_out[dst] += dinv[src]*dinv[dst]*h_in[src]; one wave per edge,
// each lane covers features {lane, lane+32}. L2-resident atomics.
__global__ __launch_bounds__(256) void edge_scatter(
    const long long* __restrict__ src, const long long* __restrict__ dst,
    const float* __restrict__ dinv, const float* __restrict__ h_in,
    float* __restrict__ h_out, int e) {
  int gid  = blockIdx.x * 256 + threadIdx.x;
  int edge = gid >> 5;
  int lane = gid & 31;
  if (edge < e) {
    long long s = src[edge];
    long long d = dst[edge];
    float w = dinv[s] * dinv[d];
    const float* hi = h_in + (size_t)s * NCLS;
    float* ho = h_out + (size_t)d * NCLS;
    atomicAdd(&ho[lane], w * hi[lane]);
    atomicAdd(&ho[lane + 32], w * hi[lane + 32]);
  }
}

// hidden = (init ? t*h : hidden + t*h), t = temp[kidx]
__global__ __launch_bounds__(256) void gpr_axpy(const float* __restrict__ h,
                                                const float* __restrict__ temp,
                                                int kidx, float* __restrict__ hidden,
                                                int count, int init) {
  int idx = blockIdx.x * 256 + threadIdx.x;
  if (idx < count) {
    float t = temp[kidx];
    hidden[idx] = init ? t * h[idx] : fmaf(t, h[idx], hidden[idx]);
  }
}

// In-place log_softmax over 64 classes; one wave32 per row, 2 vals per lane.
__global__ __launch_bounds__(256) void log_softmax64(float* __restrict__ out, int n) {
  int gid  = blockIdx.x * 256 + threadIdx.x;
  int node = gid >> 5;
  int lane = gid & 31;
  if (node >= n) return;
  float* row = out + (size_t)node * NCLS;
  float a = row[lane];
  float b = row[lane + 32];
  float mx = fmaxf(a, b);
#pragma unroll
  for (int o = 16; o > 0; o >>= 1) mx = fmaxf(mx, __shfl_xor(mx, o, 32));
  float s = __expf(a - mx) + __expf(b - mx);
#pragma unroll
  for (int o = 16; o > 0; o >>= 1) s += __shfl_xor(s, o, 32);
  float lse = mx + __logf(s);
  row[lane] = a - lse;
  row[lane + 32] = b - lse;
}

// ---------------------------------------------------------------------------
// Launch: inputs = {x, edge_index, W1, b1, W2, b2, temp}; output = [N,64] f32.
// Workspace layout (floats): h_pp0[N*64] | h_pp1[N*64] | dinv[N]
// hidden accumulates directly in d_out.
// ---------------------------------------------------------------------------
extern "C" void kernel_launch(void* const* d_in, const int* in_sizes, int n_in,
                              void* d_out, int out_size, void* d_ws, size_t ws_size,
                              hipStream_t stream) {
  const float* x  = (const float*)d_in[0];
  const long long* ei = (const long long*)d_in[1];  // int64 [2, E]
  const float* W1 = (const float*)d_in[2];
  const float* b1 = (const float*)d_in[3];
  const float* W2 = (const float*)d_in[4];
  const float* b2 = (const float*)d_in[5];
  const float* temp = (const float*)d_in[6];

  const int n = in_sizes[0] / NFEAT;
  const int e = in_sizes[1] / 2;
  const long long* src = ei;
  const long long* dst = ei + e;

  float* h0   = (float*)d_ws;                    // ping
  float* h1   = h0 + (size_t)n * NCLS;           // pong
  float* dinv = h1 + (size_t)n * NCLS;
  float* hidden = (float*)d_out;

  const int nb256 = (n + 255) / 256;
  const int feat_blocks = (n * NCLS + 255) / 256;

  // 1) Fused MLP with fp32 WMMA.
  gpr_mlp_wmma<<<(n + 15) / 16, 256, 0, stream>>>(x, W1, b1, W2, b2, h0, n);

  // 2) GCN normalization.
  deg_init<<<nb256, 256, 0, stream>>>(dinv, n);
  deg_accum<<<(e + 255) / 256, 256, 0, stream>>>(dst, dinv, e);
  deg_to_dinv<<<nb256, 256, 0, stream>>>(dinv, n);

  // 3) hidden = temp[0] * h0
  gpr_axpy<<<feat_blocks, 256, 0, stream>>>(h0, temp, 0, hidden, n * NCLS, 1);

  // 4) K propagation hops (ping-pong h buffers, L2-resident scatter).
  float* h_in = h0;
  float* h_out = h1;
  const int scat_blocks = (e * 32 + 255) / 256;
  for (int k = 0; k < KHOPS; ++k) {
    hop_init<<<feat_blocks, 256, 0, stream>>>(h_in, dinv, h_out, n);
    edge_scatter<<<scat_blocks, 256, 0, stream>>>(src, dst, dinv, h_in, h_out, e);
    gpr_axpy<<<feat_blocks, 256, 0, stream>>>(h_out, temp, k + 1, hidden, n * NCLS, 0);
    float* t = h_in; h_in = h_out; h_out = t;
  }

  // 5) log_softmax in place on d_out.
  log_softmax64<<<(n * 32 + 255) / 256, 256, 0, stream>>>(hidden, n);
}